// BiasAddLayerNormFP8_7430293422800
// MI455X (gfx1250) — compile-verified
//
#include <hip/hip_runtime.h>
#include <hip/hip_bf16.h>
#include <stdint.h>

// ---------------------------------------------------------------------------
// Fused bias-add + residual + LayerNorm + FP8(E4M3) quant + amax for
// [32768 rows x 1024] fp32. Pure bandwidth problem (~416MB @ 23.3TB/s).
// CDNA5 path: async global->LDS DMA (ASYNCcnt, saddr form + inst offset so
// issue cost is ~0 VALU), double-buffered tiles, wave32 butterfly reductions,
// v_cvt_pk_fp8_f32 packing, non-temporal loads/stores (data touched once).
// ---------------------------------------------------------------------------

typedef float v4f __attribute__((ext_vector_type(4)));

constexpr int   HDIM            = 1024;            // hidden size
constexpr int   ROWS_PER_TILE   = 8;               // one row per wave32
constexpr int   THREADS         = 256;             // 8 waves
constexpr int   TILES_PER_BLOCK = 8;
constexpr int   PARAM_BYTES     = 3 * HDIM * 4;    // bias | gamma | beta = 12KB
constexpr int   XBUF_BYTES      = ROWS_PER_TILE * HDIM * 4;    // 32KB
constexpr int   BUF_BYTES       = 2 * XBUF_BYTES;              // x + residual = 64KB
constexpr int   SMEM_BYTES      = PARAM_BYTES + 2 * BUF_BYTES; // 140KB (double buffer)
constexpr float EPSF            = 1e-5f;
constexpr float FP8_E4M3_MAXF   = 448.0f;

// Async DMA, GVS mode: copies 16B from global SGPR base + VGPR offset + imm to the LDS address in DST + imm.
// INST_OFFSET is applied to BOTH sides (ISA 08 §4.4), so the k*4096 tile
// stride lives in the immediate: zero per-issue VALU address math.
#define ASYNC_B128_NT(DST, VOFF, SBASE, OFF)                                     \
    asm volatile("global_load_async_to_lds_b128 %0, %1, %2 offset:" #OFF         \
                 " th:TH_LOAD_NT"                                                \
                 :: "v"(DST), "v"(VOFF), "s"(SBASE) : "memory")

#define ASYNC_TILE_HALF(DST, VOFF, SBASE)      \
    do {                                       \
        ASYNC_B128_NT(DST, VOFF, SBASE, 0);    \
        ASYNC_B128_NT(DST, VOFF, SBASE, 4096); \
        ASYNC_B128_NT(DST, VOFF, SBASE, 8192); \
        ASYNC_B128_NT(DST, VOFF, SBASE, 12288);\
        ASYNC_B128_NT(DST, VOFF, SBASE, 16384);\
        ASYNC_B128_NT(DST, VOFF, SBASE, 20480);\
        ASYNC_B128_NT(DST, VOFF, SBASE, 24576);\
        ASYNC_B128_NT(DST, VOFF, SBASE, 28672);\
    } while (0)

__global__ void zero_amax_kernel(uint32_t* p) { *p = 0u; }

__global__ __launch_bounds__(THREADS) void fused_bda_ln_fp8_kernel(
    const float* __restrict__ x,
    const float* __restrict__ bias,
    const float* __restrict__ residual,
    const float* __restrict__ gamma,
    const float* __restrict__ beta,
    float*       __restrict__ out_bda,
    uint32_t*    __restrict__ out_fp8,     // 4 packed e4m3 per dword
    uint32_t*    __restrict__ amax_bits)
{
    extern __shared__ __align__(16) char smem[];

    const int tid  = threadIdx.x;
    const int lane = tid & 31;
    const int wv   = tid >> 5;

    // Flat LDS address truncates to the LDS byte offset (ISA 10.2 aperture rule).
    const uint32_t lds_base = (uint32_t)(uintptr_t)smem;
    const uint32_t voff     = (uint32_t)tid * 16u;   // per-lane 16B chunk offset

    // ---- stage per-H params into LDS once (1 float4 per thread per array) ----
    {
        v4f* sbv  = (v4f*)(smem);
        v4f* sgv  = (v4f*)(smem + HDIM * 4);
        v4f* sbtv = (v4f*)(smem + 2 * HDIM * 4);
        sbv[tid]  = ((const v4f*)bias)[tid];
        sgv[tid]  = ((const v4f*)gamma)[tid];
        sbtv[tid] = ((const v4f*)beta)[tid];
    }

    // Issue one tile's async global->LDS DMA: 8 rows each of x and residual.
    // 16 wave-instructions per wave (ASYNCcnt += 16); all address variance is
    // in SGPR base / instruction offset.
    auto issue_tile = [&](int tile, int bufsel) {
        const size_t   elemBase = (size_t)tile * (ROWS_PER_TILE * HDIM);
        const uint64_t xg = (uint64_t)(uintptr_t)(x + elemBase);
        const uint64_t rg = (uint64_t)(uintptr_t)(residual + elemBase);
        const uint32_t dx = lds_base + (uint32_t)PARAM_BYTES
                          + (uint32_t)(bufsel * BUF_BYTES) + voff;
        const uint32_t dr = dx + (uint32_t)XBUF_BYTES;
        ASYNC_TILE_HALF(dx, voff, xg);
        ASYNC_TILE_HALF(dr, voff, rg);
    };

    const int tile0 = blockIdx.x * TILES_PER_BLOCK;
    issue_tile(tile0, 0);

    const v4f* sb  = (const v4f*)(smem);
    const v4f* sg  = (const v4f*)(smem + HDIM * 4);
    const v4f* sbt = (const v4f*)(smem + 2 * HDIM * 4);

    __syncthreads();   // params visible

    float lam = 0.f;   // per-lane amax across all tiles this wave processes

    for (int t = 0; t < TILES_PER_BLOCK; ++t) {
        const int tile = tile0 + t;

        if (t + 1 < TILES_PER_BLOCK) {
            issue_tile(tile + 1, (t + 1) & 1);
            // <=16 outstanding = only next tile's DMA; current tile complete.
            asm volatile("s_wait_asynccnt 0x10" ::: "memory");
        } else {
            asm volatile("s_wait_asynccnt 0x0" ::: "memory");
        }
        __syncthreads();   // all waves' DMA for tile t visible

        const char* buf  = smem + PARAM_BYTES + (t & 1) * BUF_BYTES;
        const v4f*  xrow = (const v4f*)(buf + wv * (HDIM * 4));
        const v4f*  rrow = (const v4f*)(buf + XBUF_BYTES + wv * (HDIM * 4));
        const int   row  = tile * ROWS_PER_TILE + wv;

        // ---- pass 1: bda = residual + x + bias, accumulate sum / sumsq ----
        v4f   vv[8];
        float s = 0.f, ss = 0.f;
#pragma unroll
        for (int j = 0; j < 8; ++j) {
            const int idx = j * 32 + lane;          // lane-consecutive 16B chunks
            v4f a  = xrow[idx];
            v4f r  = rrow[idx];
            v4f b  = sb[idx];
            v4f tv = a + b + r;
            vv[j] = tv;
            s  += tv.x + tv.y + tv.z + tv.w;
            ss += tv.x * tv.x + tv.y * tv.y + tv.z * tv.z + tv.w * tv.w;
        }
        // wave32 butterfly: every lane ends with full-row sums
#pragma unroll
        for (int off = 16; off > 0; off >>= 1) {
            s  += __shfl_xor(s,  off, 32);
            ss += __shfl_xor(ss, off, 32);
        }
        const float mean = s * (1.0f / HDIM);
        const float var  = ss * (1.0f / HDIM) - mean * mean;
        const float rstd = rsqrtf(var + EPSF);

        // ---- pass 2: write bda, normalize, quantize ----
        v4f*      orow = (v4f*)out_bda + (size_t)row * (HDIM / 4);
        uint32_t* frow = out_fp8 + (size_t)row * (HDIM / 4);
#pragma unroll
        for (int j = 0; j < 8; ++j) {
            const int idx = j * 32 + lane;
            v4f tv = vv[j];
            __builtin_nontemporal_store(tv, &orow[idx]);   // streamed out once

            v4f g  = sg[idx];
            v4f be = sbt[idx];
            v4f ln = (tv - mean) * rstd * g + be;

            lam = fmaxf(lam, fmaxf(fmaxf(fabsf(ln.x), fabsf(ln.y)),
                                   fmaxf(fabsf(ln.z), fabsf(ln.w))));

            ln.x = fminf(fmaxf(ln.x, -FP8_E4M3_MAXF), FP8_E4M3_MAXF);
            ln.y = fminf(fmaxf(ln.y, -FP8_E4M3_MAXF), FP8_E4M3_MAXF);
            ln.z = fminf(fmaxf(ln.z, -FP8_E4M3_MAXF), FP8_E4M3_MAXF);
            ln.w = fminf(fmaxf(ln.w, -FP8_E4M3_MAXF), FP8_E4M3_MAXF);

            int p = __builtin_amdgcn_cvt_pk_fp8_f32(ln.x, ln.y, 0, false);
            p     = __builtin_amdgcn_cvt_pk_fp8_f32(ln.z, ln.w, p, true);
            __builtin_nontemporal_store((uint32_t)p, &frow[idx]);
        }
        __syncthreads();   // buffer (t&1) free for tile t+2's DMA
    }

    // one amax reduction + one atomic per wave for the whole block's work
#pragma unroll
    for (int off = 16; off > 0; off >>= 1)
        lam = fmaxf(lam, __shfl_xor(lam, off, 32));
    if (lane == 0)
        atomicMax(amax_bits, __float_as_uint(lam));   // |ln| >= 0: uint order == float order
}

extern "C" void kernel_launch(void* const* d_in, const int* in_sizes, int n_in,
                              void* d_out, int out_size, void* d_ws, size_t ws_size,
                              hipStream_t stream) {
    const float* x        = (const float*)d_in[0];
    const float* bias     = (const float*)d_in[1];
    const float* residual = (const float*)d_in[2];
    const float* gamma    = (const float*)d_in[3];
    const float* beta     = (const float*)d_in[4];

    const long long n     = (long long)in_sizes[0];      // B*S*H = 33554432
    const int       nrows = (int)(n / HDIM);             // 32768

    // d_out layout (return-order concat): bda fp32 [n] | ln_fp8 e4m3 [n] | amax fp32
    float*    out_bda = (float*)d_out;
    uint32_t* out_fp8 = (uint32_t*)((char*)d_out + (size_t)n * sizeof(float));
    uint32_t* amax    = (uint32_t*)((char*)d_out + (size_t)n * sizeof(float) + (size_t)n);

    zero_amax_kernel<<<1, 1, 0, stream>>>(amax);

    const int tiles  = nrows / ROWS_PER_TILE;            // 4096
    const int blocks = tiles / TILES_PER_BLOCK;          // 512
    fused_bda_ln_fp8_kernel<<<blocks, THREADS, SMEM_BYTES, stream>>>(
        x, bias, residual, gamma, beta, out_bda, out_fp8, amax);
}